// WanRotaryPosEmbedS2VStyle_84430467104974
// MI455X (gfx1250) — compile-verified
//
#include <hip/hip_runtime.h>
#include <stdint.h>

// ---- problem constants (from reference: HEAD_DIM=128, PATCH=(1,2,2), hs 120x120) ----
#define T_DIMC   44
#define H_DIMC   42
#define W_DIMC   42
#define HEIGHT   60
#define WIDTH    60
#define FIXED_REF 30
#define TROWS    64                      // max staged t rows (video rows + 1 ref row)

// ---- LDS layout (float offsets) ----
#define OFF_TC  0
#define OFF_TS  (TROWS * T_DIMC)                  // 2816
#define OFF_HC  (2 * TROWS * T_DIMC)              // 5632
#define OFF_HS  (OFF_HC + HEIGHT * H_DIMC)        // 8152
#define OFF_WC  (OFF_HS + HEIGHT * H_DIMC)        // 10672
#define OFF_WS  (OFF_WC + WIDTH * W_DIMC)         // 13192
#define SM_FLOATS (OFF_WS + WIDTH * W_DIMC)       // 15712 floats = 62848 B

typedef __attribute__((ext_vector_type(4))) unsigned int v4u;
typedef __attribute__((ext_vector_type(8))) int v8i;
typedef __attribute__((ext_vector_type(4))) int v4i;

#if defined(__HIP_DEVICE_COMPILE__) && __has_builtin(__builtin_amdgcn_tensor_load_to_lds) && __has_builtin(__builtin_amdgcn_s_wait_tensorcnt)
#define USE_TDM 1
#else
#define USE_TDM 0
#endif

#if USE_TDM
// Issue one TDM 2D-tile load: tile_w elems/row (f32), tile_h rows, row stride in elems.
// D# packing per CDNA5 ISA.
__device__ __forceinline__ void tdm_load_2d(const float* gsrc, unsigned lds_byte_off,
                                            unsigned tile_w, unsigned tile_h,
                                            unsigned stride_elems) {
  unsigned long long ga = (unsigned long long)(uintptr_t)gsrc;
  v4u g0 = {
      1u,                                                    // count=1, user descriptor
      lds_byte_off,                                          // lds_addr (bytes)
      (unsigned)ga,                                          // global_addr[31:0]
      (unsigned)((ga >> 32) & 0x01FFFFFFull) | (2u << 30)    // global_addr[56:32] | type=2
  };
  v8i g1 = {
      (int)(2u << 16),                                       // data_size = 4 bytes
      (int)((tile_w & 0xFFFFu) << 16),                       // tensor_dim0[15:0] @ [31:16]
      (int)((tile_w >> 16) | ((tile_h & 0xFFFFu) << 16)),    // t_dim0[31:16] | t_dim1[15:0]
      (int)((tile_h >> 16) | ((tile_w & 0xFFFFu) << 16)),    // t_dim1[31:16] | tile_dim0
      (int)(tile_h & 0xFFFFu),                               // tile_dim1 | tile_dim2=0
      (int)stride_elems,                                     // tensor_dim0_stride[31:0]
      0, 0                                                   // stride0 hi / dim1_stride (unused 2D)
  };
  v4i gz = {0, 0, 0, 0};
#if __clang_major__ >= 23
  v8i gz8 = {0, 0, 0, 0, 0, 0, 0, 0};
  __builtin_amdgcn_tensor_load_to_lds(g0, g1, gz, gz, gz8, 0);
#else
  __builtin_amdgcn_tensor_load_to_lds(g0, g1, gz, gz, 0);
#endif
}
#endif

__global__ __launch_bounds__(256) void wan_rope_s2v_kernel(
    const float* __restrict__ t_cos, const float* __restrict__ h_cos,
    const float* __restrict__ w_cos, const float* __restrict__ t_sin,
    const float* __restrict__ h_sin, const float* __restrict__ w_sin,
    const int* __restrict__ p_nvf, float* __restrict__ out, int P) {
  __shared__ float sm[SM_FLOATS];

  int vpp = p_nvf[0];                 // p_t == 1 -> video_pp = num_video_frames
  if (vpp > TROWS - 1) vpp = TROWS - 1;
  if (vpp < 0) vpp = 0;

#if USE_TDM
  if (threadIdx.x < 32) {             // wave 0 drives the tensor DMA for this block
    unsigned base = (unsigned)(uintptr_t)&sm[0];   // low 32 bits of flat LDS addr = LDS offset
    tdm_load_2d(t_cos,                      base + OFF_TC * 4u,                T_DIMC, (unsigned)vpp, T_DIMC);
    tdm_load_2d(t_cos + FIXED_REF * T_DIMC, base + (OFF_TC + vpp * T_DIMC) * 4u, T_DIMC, 1u, T_DIMC);
    tdm_load_2d(t_sin,                      base + OFF_TS * 4u,                T_DIMC, (unsigned)vpp, T_DIMC);
    tdm_load_2d(t_sin + FIXED_REF * T_DIMC, base + (OFF_TS + vpp * T_DIMC) * 4u, T_DIMC, 1u, T_DIMC);
    tdm_load_2d(h_cos, base + OFF_HC * 4u, H_DIMC, HEIGHT, H_DIMC);
    tdm_load_2d(h_sin, base + OFF_HS * 4u, H_DIMC, HEIGHT, H_DIMC);
    tdm_load_2d(w_cos, base + OFF_WC * 4u, W_DIMC, WIDTH,  W_DIMC);
    tdm_load_2d(w_sin, base + OFF_WS * 4u, W_DIMC, WIDTH,  W_DIMC);
    __builtin_amdgcn_s_wait_tensorcnt(0);
  }
  __syncthreads();
#else
  // Fallback staging via plain loads (only taken if the TDM builtin is unavailable).
  for (int i = threadIdx.x; i < (vpp + 1) * T_DIMC; i += blockDim.x) {
    int r = i / T_DIMC, c = i - r * T_DIMC;
    int sr = (r < vpp) ? r : FIXED_REF;
    sm[OFF_TC + i] = t_cos[sr * T_DIMC + c];
    sm[OFF_TS + i] = t_sin[sr * T_DIMC + c];
  }
  for (int i = threadIdx.x; i < HEIGHT * H_DIMC; i += blockDim.x) {
    sm[OFF_HC + i] = h_cos[i];
    sm[OFF_HS + i] = h_sin[i];
    sm[OFF_WC + i] = w_cos[i];
    sm[OFF_WS + i] = w_sin[i];
  }
  __syncthreads();
#endif

  // Wave-per-position streaming stores: lane l owns channels [4l, 4l+4).
  const int lane = threadIdx.x & 31;
  const int wid  = (int)((blockIdx.x * blockDim.x + threadIdx.x) >> 5);
  const int nw   = (int)((gridDim.x * blockDim.x) >> 5);
  const int c0   = lane << 2;

  float* __restrict__ out_cos = out;
  float* __restrict__ out_sin = out + (long)P * 128;

  for (int pos = wid; pos < P; pos += nw) {
    int f   = pos / (HEIGHT * WIDTH);
    int rem = pos - f * (HEIGHT * WIDTH);
    int y   = rem / WIDTH;
    int x   = rem - y * WIDTH;
    int tr  = (f < vpp) ? f : vpp;     // ref frames all map to staged FIXED_REF row

    const float* tc = sm + OFF_TC + tr * T_DIMC;
    const float* ts = sm + OFF_TS + tr * T_DIMC;
    const float* hc = sm + OFF_HC + y * H_DIMC;
    const float* hs = sm + OFF_HS + y * H_DIMC;
    const float* wc = sm + OFF_WC + x * W_DIMC;
    const float* ws = sm + OFF_WS + x * W_DIMC;

    float cv[4], sv[4];
#pragma unroll
    for (int j = 0; j < 4; ++j) {
      int ch = c0 + j;
      float a, b;
      if (ch < T_DIMC)                { a = tc[ch];            b = ts[ch]; }
      else if (ch < T_DIMC + H_DIMC)  { a = hc[ch - T_DIMC];   b = hs[ch - T_DIMC]; }
      else                            { a = wc[ch - 86];       b = ws[ch - 86]; }
      cv[j] = a; sv[j] = b;
    }
    float4 vcos = make_float4(cv[0], cv[1], cv[2], cv[3]);
    float4 vsin = make_float4(sv[0], sv[1], sv[2], sv[3]);
    *(float4*)(out_cos + (long)pos * 128 + c0) = vcos;   // global_store_b128
    *(float4*)(out_sin + (long)pos * 128 + c0) = vsin;   // global_store_b128
  }
}

extern "C" void kernel_launch(void* const* d_in, const int* in_sizes, int n_in,
                              void* d_out, int out_size, void* d_ws, size_t ws_size,
                              hipStream_t stream) {
  (void)in_sizes; (void)n_in; (void)d_ws; (void)ws_size;
  const float* t_cos = (const float*)d_in[1];
  const float* h_cos = (const float*)d_in[2];
  const float* w_cos = (const float*)d_in[3];
  const float* t_sin = (const float*)d_in[4];
  const float* h_sin = (const float*)d_in[5];
  const float* w_sin = (const float*)d_in[6];
  const int*   p_nvf = (const int*)d_in[7];
  float* out = (float*)d_out;

  // out = concat(cos, sin), each P x 128 -> P positions
  int P = out_size / (2 * 128);

  dim3 block(256);
  dim3 grid(1280);
  hipLaunchKernelGGL(wan_rope_s2v_kernel, grid, block, 0, stream,
                     t_cos, h_cos, w_cos, t_sin, h_sin, w_sin, p_nvf, out, P);
}